// EMALayer_39084202393603
// MI455X (gfx1250) — compile-verified
//
#include <hip/hip_runtime.h>

// N=8192, D=32, H=128, Z=32, V=32
#define NN 8192

typedef __attribute__((ext_vector_type(2))) float v2f;
typedef __attribute__((ext_vector_type(8))) float v8f;

#define WMMA_F32(A, B, C) \
  __builtin_amdgcn_wmma_f32_16x16x4_f32(false, (A), false, (B), (short)0, (C), false, false)

// ---------------------------------------------------------------------------
// c[d] = dot(beta[d,:], mu[d,:]) over H=128
// ---------------------------------------------------------------------------
__global__ void cdot_kernel(const float* __restrict__ beta,
                            const float* __restrict__ mu,
                            float* __restrict__ c) {
  int d = threadIdx.x;  // 32 threads
  float acc = 0.f;
  for (int h = 0; h < 128; ++h) acc += beta[d * 128 + h] * mu[d * 128 + h];
  c[d] = acc;
}

// ---------------------------------------------------------------------------
// Column-wise inclusive prefix sum of x [N,32]; one block per column.
// ---------------------------------------------------------------------------
__global__ void colscan_kernel(const float* __restrict__ x,
                               float* __restrict__ prefix) {
  __shared__ float s[256];
  const int d = blockIdx.x;
  const int t = threadIdx.x;
  float carry = 0.f;
  for (int base = 0; base < NN; base += 256) {
    s[t] = x[(base + t) * 32 + d];
    __syncthreads();
    for (int off = 1; off < 256; off <<= 1) {
      float add = (t >= off) ? s[t - off] : 0.f;
      __syncthreads();
      s[t] += add;
      __syncthreads();
    }
    prefix[(base + t) * 32 + d] = s[t] + carry;
    float tot = s[255];
    __syncthreads();
    carry += tot;
  }
}

// ---------------------------------------------------------------------------
// z = prefix*(c)·W_z + b_z ; Q = z*k_q+mu_q ; K = z*k_k+mu_k ; V = x·W_v + b_v
// One thread per (row, col). 8 rows per 256-thread block.
// ---------------------------------------------------------------------------
__global__ void qkv_kernel(const float* __restrict__ x,
                           const float* __restrict__ prefix,
                           const float* __restrict__ c,
                           const float* __restrict__ W_z, const float* __restrict__ b_z,
                           const float* __restrict__ W_v, const float* __restrict__ b_v,
                           const float* __restrict__ k_q, const float* __restrict__ mu_q,
                           const float* __restrict__ k_k, const float* __restrict__ mu_k,
                           float* __restrict__ Q, float* __restrict__ K,
                           float* __restrict__ V) {
  const int t = threadIdx.x;
  const int row = blockIdx.x * 8 + (t >> 5);
  const int col = t & 31;
  float z = b_z[col];
  float vv = b_v[col];
#pragma unroll
  for (int d = 0; d < 32; ++d) {
    z  += prefix[row * 32 + d] * c[d] * W_z[d * 32 + col];
    vv += x[row * 32 + d] * W_v[d * 32 + col];
  }
  Q[row * 32 + col] = z * k_q[col] + mu_q[col];
  K[row * 32 + col] = z * k_k[col] + mu_k[col];
  V[row * 32 + col] = vv;
}

// ---------------------------------------------------------------------------
// Flash attention (headdim 32, N=8192) + fused output projection O @ n.
// 128 threads = 4 waves; each wave owns 16 query rows.
// f32 WMMA 16x16x4 throughout; K/V tiles of 64 keys staged via LDS.
// Softmax state is updated once per 64-key tile (deferred reduction); both
// matmul phases run 4 independent WMMA accumulator chains to hide the
// WMMA->WMMA dependency latency.
// ---------------------------------------------------------------------------
__launch_bounds__(128)
__global__ void flash_kernel(const float* __restrict__ Q,
                             const float* __restrict__ Kg,
                             const float* __restrict__ Vg,
                             const float* __restrict__ nmat,   // [32,128]
                             float* __restrict__ out) {        // [N,128]
  __shared__ float Ktile[64 * 32];
  __shared__ float Vtile[64 * 32];
  __shared__ float Pbuf[4][16 * 64];

  const int tid  = threadIdx.x;
  const int wave = tid >> 5;
  const int lane = tid & 31;
  const int hi   = lane >> 4;   // 0: rows r, 1: rows r+8 of C layout
  const int ln   = lane & 15;
  const int q0   = blockIdx.x * 64 + wave * 16;

  // Preload Q as 8 A-fragments (16x4 each, K = 0..31).
  v2f qa[8];
#pragma unroll
  for (int j = 0; j < 8; ++j) {
    const int ks = 4 * j + 2 * hi;
    const float* p = Q + (q0 + ln) * 32 + ks;
    qa[j].x = p[0];
    qa[j].y = p[1];
  }

  float m_run[8], l_run[8];
  const v8f vzero = {0.f, 0.f, 0.f, 0.f, 0.f, 0.f, 0.f, 0.f};
  v8f o0a = vzero, o0b = vzero, o1a = vzero, o1b = vzero;
#pragma unroll
  for (int r = 0; r < 8; ++r) { m_run[r] = -1e30f; l_run[r] = 0.f; }

  for (int kb = 0; kb < NN / 64; ++kb) {
    __syncthreads();
    {
      const float4* ks4 = (const float4*)(Kg + kb * 2048);
      const float4* vs4 = (const float4*)(Vg + kb * 2048);
      float4* kd4 = (float4*)Ktile;
      float4* vd4 = (float4*)Vtile;
#pragma unroll
      for (int i = 0; i < 4; ++i) {
        kd4[tid + i * 128] = ks4[tid + i * 128];
        vd4[tid + i * 128] = vs4[tid + i * 128];
      }
      if (kb + 1 < NN / 64) {   // hint next tiles toward the caches
        __builtin_prefetch(Kg + (kb + 1) * 2048 + tid * 16, 0, 1);
        __builtin_prefetch(Vg + (kb + 1) * 2048 + tid * 16, 0, 1);
      }
    }
    __syncthreads();

    // ---- S = Q · K^T for all four 16-key chunks (32 WMMAs, 4 chains) ----
    v8f s[4];
#pragma unroll
    for (int sub = 0; sub < 4; ++sub) s[sub] = vzero;
#pragma unroll
    for (int j = 0; j < 8; ++j) {
      const int ks = 4 * j + 2 * hi;
#pragma unroll
      for (int sub = 0; sub < 4; ++sub) {
        const float* kt = Ktile + sub * 16 * 32;
        v2f b;
        b.x = kt[ln * 32 + ks];
        b.y = kt[ln * 32 + ks + 1];
        s[sub] = WMMA_F32(qa[j], b, s[sub]);
      }
    }

    // ---- Deferred streaming-softmax update over the whole 64-key tile ----
    float mn[8], alpha[8];
#pragma unroll
    for (int r = 0; r < 8; ++r) {
      float mx = fmaxf(fmaxf(s[0][r], s[1][r]), fmaxf(s[2][r], s[3][r]));
      mx = fmaxf(mx, __shfl_xor(mx, 8, 16));
      mx = fmaxf(mx, __shfl_xor(mx, 4, 16));
      mx = fmaxf(mx, __shfl_xor(mx, 2, 16));
      mx = fmaxf(mx, __shfl_xor(mx, 1, 16));
      const float m = fmaxf(m_run[r], mx);
      alpha[r] = __expf(m_run[r] - m);
      m_run[r] = m;
      mn[r] = m;
    }
#pragma unroll
    for (int r = 0; r < 8; ++r) {
      float sum = 0.f;
#pragma unroll
      for (int sub = 0; sub < 4; ++sub) {
        const float p = __expf(s[sub][r] - mn[r]);
        s[sub][r] = p;
        sum += p;
      }
      sum += __shfl_xor(sum, 8, 16);
      sum += __shfl_xor(sum, 4, 16);
      sum += __shfl_xor(sum, 2, 16);
      sum += __shfl_xor(sum, 1, 16);
      l_run[r] = l_run[r] * alpha[r] + sum;
      o0a[r] *= alpha[r];
      o0b[r] *= alpha[r];
      o1a[r] *= alpha[r];
      o1b[r] *= alpha[r];
    }

    // ---- C-layout -> A-layout for P (16x64) via per-wave LDS scratch ----
    float* pb = Pbuf[wave];
#pragma unroll
    for (int sub = 0; sub < 4; ++sub) {
#pragma unroll
      for (int r = 0; r < 8; ++r)
        pb[(r + 8 * hi) * 64 + sub * 16 + ln] = s[sub][r];
    }
    asm volatile("s_wait_dscnt 0" ::: "memory");

    // ---- O += P · V over K-dim 64 (32 WMMAs, 4 independent chains) ----
#pragma unroll
    for (int jj = 0; jj < 8; ++jj) {
      const int ks0 = 8 * jj + 2 * hi;       // even fragment
      const int ks1 = 8 * jj + 4 + 2 * hi;   // odd fragment
      v2f pa0, pa1;
      pa0.x = pb[ln * 64 + ks0];
      pa0.y = pb[ln * 64 + ks0 + 1];
      pa1.x = pb[ln * 64 + ks1];
      pa1.y = pb[ln * 64 + ks1 + 1];
      v2f vb00, vb01, vb10, vb11;
      vb00.x = Vtile[ks0 * 32 + ln];
      vb00.y = Vtile[(ks0 + 1) * 32 + ln];
      vb01.x = Vtile[ks0 * 32 + 16 + ln];
      vb01.y = Vtile[(ks0 + 1) * 32 + 16 + ln];
      vb10.x = Vtile[ks1 * 32 + ln];
      vb10.y = Vtile[(ks1 + 1) * 32 + ln];
      vb11.x = Vtile[ks1 * 32 + 16 + ln];
      vb11.y = Vtile[(ks1 + 1) * 32 + 16 + ln];
      o0a = WMMA_F32(pa0, vb00, o0a);
      o1a = WMMA_F32(pa0, vb01, o1a);
      o0b = WMMA_F32(pa1, vb10, o0b);
      o1b = WMMA_F32(pa1, vb11, o1b);
    }
  }

  // Merge partial accumulators and normalize by the softmax denominator.
  v8f o0, o1;
#pragma unroll
  for (int r = 0; r < 8; ++r) {
    const float inv = 1.0f / l_run[r];
    o0[r] = (o0a[r] + o0b[r]) * inv;
    o1[r] = (o1a[r] + o1b[r]) * inv;
  }

  // Epilogue: out_tile[16,128] = O[16,32] @ nmat[32,128], fused via WMMA.
  float* pb = Pbuf[wave];
#pragma unroll
  for (int r = 0; r < 8; ++r) {
    pb[(r + 8 * hi) * 32 + ln]      = o0[r];
    pb[(r + 8 * hi) * 32 + 16 + ln] = o1[r];
  }
  asm volatile("s_wait_dscnt 0" ::: "memory");

  v2f oa[8];
#pragma unroll
  for (int j = 0; j < 8; ++j) {
    const int ks = 4 * j + 2 * hi;
    oa[j].x = pb[ln * 32 + ks];
    oa[j].y = pb[ln * 32 + ks + 1];
  }
#pragma unroll
  for (int ct = 0; ct < 8; ++ct) {
    v8f acc = vzero;
#pragma unroll
    for (int j = 0; j < 8; ++j) {
      const int ks = 4 * j + 2 * hi;
      v2f b;
      b.x = nmat[ks * 128 + ct * 16 + ln];
      b.y = nmat[(ks + 1) * 128 + ct * 16 + ln];
      acc = WMMA_F32(oa[j], b, acc);
    }
#pragma unroll
    for (int r = 0; r < 8; ++r) {
      out[(q0 + r + 8 * hi) * 128 + ct * 16 + ln] = acc[r];
    }
  }
}

// ---------------------------------------------------------------------------
extern "C" void kernel_launch(void* const* d_in, const int* in_sizes, int n_in,
                              void* d_out, int out_size, void* d_ws, size_t ws_size,
                              hipStream_t stream) {
  const float* x    = (const float*)d_in[0];
  const float* beta = (const float*)d_in[1];
  const float* mu   = (const float*)d_in[2];
  const float* nmat = (const float*)d_in[3];
  const float* W_z  = (const float*)d_in[4];
  const float* b_z  = (const float*)d_in[5];
  const float* W_v  = (const float*)d_in[6];
  const float* b_v  = (const float*)d_in[7];
  const float* k_q  = (const float*)d_in[8];
  const float* mu_q = (const float*)d_in[9];
  const float* k_k  = (const float*)d_in[10];
  const float* mu_k = (const float*)d_in[11];

  // Workspace layout (floats): c[32] (padded to 64) | prefix | Q | K | V
  float* ws     = (float*)d_ws;
  float* c      = ws;
  float* prefix = ws + 64;
  float* Qb     = prefix + NN * 32;
  float* Kb     = Qb + NN * 32;
  float* Vb     = Kb + NN * 32;

  cdot_kernel<<<1, 32, 0, stream>>>(beta, mu, c);
  colscan_kernel<<<32, 256, 0, stream>>>(x, prefix);
  qkv_kernel<<<NN / 8, 256, 0, stream>>>(x, prefix, c, W_z, b_z, W_v, b_v,
                                         k_q, mu_q, k_k, mu_k, Qb, Kb, Vb);
  flash_kernel<<<NN / 64, 128, 0, stream>>>(Qb, Kb, Vb, nmat, (float*)d_out);
}